// ComputeAllAtomCoords_34153579938258
// MI455X (gfx1250) — compile-verified
//
#include <hip/hip_runtime.h>
#include <hip/hip_bf16.h>
#include <stdint.h>

#define NAA 80
#define NATOMS 36

typedef int v4i __attribute__((ext_vector_type(4)));
typedef __attribute__((address_space(1))) v4i* as1_v4i;
typedef __attribute__((address_space(3))) v4i* as3_v4i;

// ---------------------------------------------------------------------------
// gfx1250 async global->LDS staging (tracked by ASYNCcnt)
// ---------------------------------------------------------------------------
#if defined(__AMDGCN__) && __has_builtin(__builtin_amdgcn_global_load_async_to_lds_b128)
#define USE_ASYNC_B128 1
#endif

__device__ __forceinline__ void ldsAsyncCopy16(const float* g, float* l) {
#if defined(USE_ASYNC_B128)
    __builtin_amdgcn_global_load_async_to_lds_b128(
        (as1_v4i)(uintptr_t)g,   // global address (flat == global numerically)
        (as3_v4i)l,              // addrspacecast generic->LDS
        /*offset=*/0, /*cpol=*/0);
#else
    *reinterpret_cast<float4*>(l) = *reinterpret_cast<const float4*>(g);
#endif
}

__device__ __forceinline__ void ldsAsyncWait() {
#if defined(__AMDGCN__)
#if __has_builtin(__builtin_amdgcn_s_wait_asynccnt)
    __builtin_amdgcn_s_wait_asynccnt(0);
#else
    asm volatile("s_wait_asynccnt 0" ::: "memory");
#endif
#endif
}

// ---------------------------------------------------------------------------
// small matrix helpers (row-major 4x4)
// ---------------------------------------------------------------------------
__device__ __forceinline__ void mm4(const float* A, const float* B, float* C) {
#pragma unroll
    for (int i = 0; i < 4; ++i) {
#pragma unroll
        for (int j = 0; j < 4; ++j) {
            float acc = A[i * 4 + 0] * B[0 * 4 + j];
            acc = fmaf(A[i * 4 + 1], B[1 * 4 + j], acc);
            acc = fmaf(A[i * 4 + 2], B[2 * 4 + j], acc);
            acc = fmaf(A[i * 4 + 3], B[3 * 4 + j], acc);
            C[i * 4 + j] = acc;
        }
    }
}

// C = M @ rotX(c,s): only columns 1,2 change
__device__ __forceinline__ void mulRotX(const float* M, float c, float s, float* C) {
#pragma unroll
    for (int i = 0; i < 4; ++i) {
        float m1 = M[i * 4 + 1], m2 = M[i * 4 + 2];
        C[i * 4 + 0] = M[i * 4 + 0];
        C[i * 4 + 1] = fmaf(m1, c, m2 * s);
        C[i * 4 + 2] = fmaf(m2, c, -m1 * s);
        C[i * 4 + 3] = M[i * 4 + 3];
    }
}

// C = M @ rotZ(c,s): only columns 0,1 change
__device__ __forceinline__ void mulRotZ(const float* M, float c, float s, float* C) {
#pragma unroll
    for (int i = 0; i < 4; ++i) {
        float m0 = M[i * 4 + 0], m1 = M[i * 4 + 1];
        C[i * 4 + 0] = fmaf(m0, c, m1 * s);
        C[i * 4 + 1] = fmaf(m1, c, -m0 * s);
        C[i * 4 + 2] = M[i * 4 + 2];
        C[i * 4 + 3] = M[i * 4 + 3];
    }
}

// Q = 3x3 rotation about axis u with cos=ct, sin=st (row-major, 9 floats)
__device__ __forceinline__ void buildQ(float ct, float st, float u0, float u1, float u2,
                                       float* Q) {
    float omc = 1.0f - ct;
    Q[0] = ct + u0 * u0 * omc;  Q[1] = u0 * u1 * omc - u2 * st;  Q[2] = u0 * u2 * omc + u1 * st;
    Q[3] = u0 * u1 * omc + u2 * st;  Q[4] = ct + u1 * u1 * omc;  Q[5] = u1 * u2 * omc - u0 * st;
    Q[6] = u0 * u2 * omc - u1 * st;  Q[7] = u1 * u2 * omc + u0 * st;  Q[8] = ct + u2 * u2 * omc;
}

// C = M @ [[Q,0],[0,1]] : columns 0..2 mix, column 3 copies
__device__ __forceinline__ void mulQ(const float* M, const float* Q, float* C) {
#pragma unroll
    for (int i = 0; i < 4; ++i) {
        float m0 = M[i * 4 + 0], m1 = M[i * 4 + 1], m2 = M[i * 4 + 2];
        C[i * 4 + 0] = fmaf(m0, Q[0], fmaf(m1, Q[3], m2 * Q[6]));
        C[i * 4 + 1] = fmaf(m0, Q[1], fmaf(m1, Q[4], m2 * Q[7]));
        C[i * 4 + 2] = fmaf(m0, Q[2], fmaf(m1, Q[5], m2 * Q[8]));
        C[i * 4 + 3] = M[i * 4 + 3];
    }
}

__device__ __forceinline__ void storeMat4(float* dst, const float* M) {
#pragma unroll
    for (int i = 0; i < 4; ++i)
        *reinterpret_cast<float4*>(dst + i * 4) =
            make_float4(M[i * 4 + 0], M[i * 4 + 1], M[i * 4 + 2], M[i * 4 + 3]);
}

struct F3 { float x, y, z; };   // packed 12B, align(4) -> global_store_b96

// ---------------------------------------------------------------------------
// Kernel 1: one thread per residue -> 9 frames (B,L,9,4,4)
// ---------------------------------------------------------------------------
__global__ void __launch_bounds__(256)
frames_kernel(const int* __restrict__ seq, const float* __restrict__ xyz,
              const float* __restrict__ alphas, const float* __restrict__ RTs,
              const float* __restrict__ bxyz, float* __restrict__ outF, int NR) {
    __shared__ float sRT[NAA * 7 * 16];   // 35840 B : per-AA frame tables
    __shared__ float sAX[NAA * 16];       //  5120 B : bxyz rows {0,1,2,4} per AA
    __shared__ float sXYZ[256 * 9];       //  9216 B : this block's N/Ca/C stream
    __shared__ float sALP[256 * 20];      // 20480 B : this block's alphas stream

    const int tid = threadIdx.x;
    const int blockStart = blockIdx.x * 256;
    const bool full = (blockStart + 256 <= NR);

    // ---- stage all inputs as coalesced async b128 streams ------------------
    for (int i = tid; i < NAA * 7 * 4; i += 256)        // 2240 x b128
        ldsAsyncCopy16(RTs + (size_t)i * 4, sRT + i * 4);
    for (int i = tid; i < NAA * 4; i += 256) {          // 320 x b128
        int s = i >> 2, j = i & 3;
        int srcrow = (j == 3) ? 4 : j;
        ldsAsyncCopy16(bxyz + (size_t)(s * NATOMS + srcrow) * 4, sAX + i * 4);
    }
    if (full) {
        const float* gx = xyz + (size_t)blockStart * 9;     // 16B aligned base
        for (int i = tid; i < 576; i += 256)                // 576 x b128
            ldsAsyncCopy16(gx + (size_t)i * 4, sXYZ + i * 4);
        const float* ga = alphas + (size_t)blockStart * 20; // 16B aligned base
        for (int i = tid; i < 1280; i += 256)               // 1280 x b128
            ldsAsyncCopy16(ga + (size_t)i * 4, sALP + i * 4);
    }
    ldsAsyncWait();
    __syncthreads();

    int r = blockStart + tid;
    if (r >= NR) return;

    // ---- pull this residue's inputs into registers -------------------------
    float Xr[9];
    if (full) {
#pragma unroll
        for (int k = 0; k < 9; ++k) Xr[k] = sXYZ[tid * 9 + k];   // stride 9: bank-conflict free
    } else {
#pragma unroll
        for (int k = 0; k < 9; ++k) Xr[k] = xyz[(size_t)r * 9 + k];
    }
    float Ar[20];
    if (full) {
#pragma unroll
        for (int k = 0; k < 20; ++k) Ar[k] = sALP[tid * 20 + k];
    } else {
#pragma unroll
        for (int k = 0; k < 20; ++k) Ar[k] = alphas[(size_t)r * 20 + k];
    }

    // --- rigid frame from N, Ca, C -----------------------------------------
    float Nx = Xr[0], Ny = Xr[1], Nz = Xr[2];
    float Cax = Xr[3], Cay = Xr[4], Caz = Xr[5];
    float Cx = Xr[6], Cy = Xr[7], Cz = Xr[8];
    float v1x = Cx - Cax, v1y = Cy - Cay, v1z = Cz - Caz;
    float v2x = Nx - Cax, v2y = Ny - Cay, v2z = Nz - Caz;
    float n1 = sqrtf(v1x * v1x + v1y * v1y + v1z * v1z) + 1e-8f;
    float e1x = v1x / n1, e1y = v1y / n1, e1z = v1z / n1;
    float dd = e1x * v2x + e1y * v2y + e1z * v2z;
    float u2x = v2x - e1x * dd, u2y = v2y - e1y * dd, u2z = v2z - e1z * dd;
    float n2 = sqrtf(u2x * u2x + u2y * u2y + u2z * u2z) + 1e-8f;
    float e2x = u2x / n2, e2y = u2y / n2, e2z = u2z / n2;
    float e3x = e1y * e2z - e1z * e2y;
    float e3y = e1z * e2x - e1x * e2z;
    float e3z = e1x * e2y - e1y * e2x;
    float F0[16] = {e1x, e2x, e3x, Cax,
                    e1y, e2y, e3y, Cay,
                    e1z, e2z, e3z, Caz,
                    0.f, 0.f, 0.f, 1.f};

    // --- angles -> cos/sin --------------------------------------------------
    float cs[10], sn[10];
#pragma unroll
    for (int k = 0; k < 10; ++k) {
        float a0 = Ar[2 * k], a1 = Ar[2 * k + 1];
        float n = sqrtf(a0 * a0 + a1 * a1) + 1e-6f;
        cs[k] = a0 / n;
        sn[k] = a1 / n;
    }

    int s = seq[r];
    const float* RTb = sRT + s * 112;  // 7 matrices of 16 (LDS)
    float* Fout = outF + (size_t)r * 144;

    float T1[16], T2[16];
    storeMat4(Fout + 0 * 16, F0);
    mm4(F0, RTb + 0 * 16, T1);  mulRotX(T1, cs[0], sn[0], T2);  storeMat4(Fout + 1 * 16, T2);
    mm4(F0, RTb + 1 * 16, T1);  mulRotX(T1, cs[1], sn[1], T2);  storeMat4(Fout + 2 * 16, T2);
    mm4(F0, RTb + 2 * 16, T1);  mulRotX(T1, cs[2], sn[2], T2);  storeMat4(Fout + 3 * 16, T2);

    // --- CB rotation axes from base-frame atoms (LDS) -----------------------
    const float* axd = sAX + s * 16;
    float N0x = axd[0], N0y = axd[1], N0z = axd[2];
    float CArx = axd[4], CAry = axd[5], CArz = axd[6];
    float C2x = axd[8], C2y = axd[9], C2z = axd[10];
    float CBx = axd[12], CBy = axd[13], CBz = axd[14];
    float NCrx = 0.5f * (C2x + N0x), NCry = 0.5f * (C2y + N0y), NCrz = 0.5f * (C2z + N0z);
    float bax = CBx - CArx, bay = CBy - CAry, baz = CBz - CArz;  // CBr - CAr
    float d1x = NCrx - CArx, d1y = NCry - CAry, d1z = NCrz - CArz;
    float a1x = bay * d1z - baz * d1y;
    float a1y = baz * d1x - bax * d1z;
    float a1z = bax * d1y - bay * d1x;
    float an1 = sqrtf(a1x * a1x + a1y * a1y + a1z * a1z) + 1e-8f;
    a1x /= an1; a1y /= an1; a1z /= an1;
    float NCpx = C2x - N0x, NCpy = C2y - N0y, NCpz = C2z - N0z;
    float t = (NCpx * NCrx + NCpy * NCry + NCpz * NCrz) /
              (NCrx * NCrx + NCry * NCry + NCrz * NCrz);
    float px = NCpx - t * NCrx, py = NCpy - t * NCry, pz = NCpz - t * NCrz;
    float a2x = bay * pz - baz * py;
    float a2y = baz * px - bax * pz;
    float a2z = bax * py - bay * px;
    float an2 = sqrtf(a2x * a2x + a2y * a2y + a2z * a2z) + 1e-8f;
    a2x /= an2; a2y /= an2; a2z /= an2;

    float Q1[9], Q2[9], F8[16], FA[16];
    buildQ(cs[7], sn[7], a1x, a1y, a1z, Q1);
    buildQ(cs[8], sn[8], a2x, a2y, a2z, Q2);
    mulQ(F0, Q1, T1);  mulQ(T1, Q2, F8);  storeMat4(Fout + 8 * 16, F8);

    mm4(F8, RTb + 3 * 16, T1);  mulRotX(T1, cs[3], sn[3], T2);
    mulRotZ(T2, cs[9], sn[9], FA);                 storeMat4(Fout + 4 * 16, FA);  // RTF4
    mm4(FA, RTb + 4 * 16, T1);  mulRotX(T1, cs[4], sn[4], T2);
    storeMat4(Fout + 5 * 16, T2);                                                 // RTF5
    mm4(T2, RTb + 5 * 16, T1);  mulRotX(T1, cs[5], sn[5], FA);
    storeMat4(Fout + 6 * 16, FA);                                                 // RTF6
    mm4(FA, RTb + 6 * 16, T1);  mulRotX(T1, cs[6], sn[6], T2);
    storeMat4(Fout + 7 * 16, T2);                                                 // RTF7
}

// ---------------------------------------------------------------------------
// Kernel 2: one thread per (residue, atom); frames come straight from L2
// ---------------------------------------------------------------------------
__global__ void __launch_bounds__(256)
atoms_kernel(const int* __restrict__ seq, const int* __restrict__ bidx,
             const float* __restrict__ bxyz, const float* __restrict__ frames,
             float* __restrict__ outX, int NR) {
    int g = blockIdx.x * 256 + threadIdx.x;
    int total = NR * NATOMS;
    if (g >= total) return;
    int r = g / NATOMS;
    int a = g - r * NATOMS;
    int s = seq[r];
    int idx = bidx[s * NATOMS + a];
    const float4 v = *reinterpret_cast<const float4*>(bxyz + (size_t)(s * NATOMS + a) * 4);
    const float* Fm = frames + (size_t)r * 144 + (size_t)idx * 16;
    float4 r0 = *reinterpret_cast<const float4*>(Fm + 0);
    float4 r1 = *reinterpret_cast<const float4*>(Fm + 4);
    float4 r2 = *reinterpret_cast<const float4*>(Fm + 8);
    F3 o;
    o.x = fmaf(r0.x, v.x, fmaf(r0.y, v.y, fmaf(r0.z, v.z, r0.w * v.w)));
    o.y = fmaf(r1.x, v.x, fmaf(r1.y, v.y, fmaf(r1.z, v.z, r1.w * v.w)));
    o.z = fmaf(r2.x, v.x, fmaf(r2.y, v.y, fmaf(r2.z, v.z, r2.w * v.w)));
    *reinterpret_cast<F3*>(outX + (size_t)g * 3) = o;   // single b96 store
}

extern "C" void kernel_launch(void* const* d_in, const int* in_sizes, int n_in,
                              void* d_out, int out_size, void* d_ws, size_t ws_size,
                              hipStream_t stream) {
    const int*   seq    = (const int*)d_in[0];
    const float* xyz    = (const float*)d_in[1];
    const float* alphas = (const float*)d_in[2];
    const int*   bidx   = (const int*)d_in[3];
    const float* RTs    = (const float*)d_in[4];
    const float* bxyz   = (const float*)d_in[5];
    float* out = (float*)d_out;

    const int NR = in_sizes[0];                 // B*L residues
    float* outF = out;                          // (B,L,9,4,4)
    float* outX = out + (size_t)NR * 144;       // (B,L,36,3)

    int blocks1 = (NR + 255) / 256;
    frames_kernel<<<blocks1, 256, 0, stream>>>(seq, xyz, alphas, RTs, bxyz, outF, NR);

    int total = NR * NATOMS;
    int blocks2 = (total + 255) / 256;
    atoms_kernel<<<blocks2, 256, 0, stream>>>(seq, bidx, bxyz, outF, outX, NR);
}